// SAGE_FOR_HETERO_67783173865548
// MI455X (gfx1250) — compile-verified
//
#include <hip/hip_runtime.h>
#include <hip/hip_bf16.h>

#define NN   50000
#define NE   800000
#define FDIM 128
#define KTOT 256            // fused K: [agg | x]
#define WT_PITCH 260        // LDS pitch (floats): conflict-free b64 B-fragment loads
#define MTILES (NN / 16)    // 3125, exact

typedef __attribute__((ext_vector_type(2))) float v2f;
typedef __attribute__((ext_vector_type(8))) float v8f;

__global__ void k_zero4(float4* __restrict__ p, int n4) {
  int i = blockIdx.x * blockDim.x + threadIdx.x;
  if (i < n4) p[i] = make_float4(0.f, 0.f, 0.f, 0.f);
}

__global__ void k_count(const int* __restrict__ dst, float* __restrict__ cnt) {
  int e = blockIdx.x * blockDim.x + threadIdx.x;
  if (e < NE) atomicAdd(&cnt[dst[e]], 1.0f);
}

__global__ void k_invc(const float* __restrict__ cnt, float* __restrict__ invc) {
  int i = blockIdx.x * blockDim.x + threadIdx.x;
  if (i < NN) invc[i] = 1.0f / fmaxf(cnt[i], 1.0f);
}

// One wave per edge: 32 lanes x float4 = fully coalesced 512B gather of x[src],
// scatter-add into summed[dst] with native global_atomic_add_f32.
__global__ void k_scatter(const int* __restrict__ src, const int* __restrict__ dst,
                          const float* __restrict__ x, float* __restrict__ summed) {
  int e    = blockIdx.x * 8 + (threadIdx.x >> 5);
  int lane = threadIdx.x & 31;
  if (e >= NE) return;
  int s = src[e];
  int d = dst[e];
  float4 v = ((const float4*)(x + (size_t)s * FDIM))[lane];
  float* o = summed + (size_t)d * FDIM + lane * 4;
  atomicAdd(o + 0, v.x);
  atomicAdd(o + 1, v.y);
  atomicAdd(o + 2, v.z);
  atomicAdd(o + 3, v.w);
}

// out = relu?( (summed .* invc) @ Wl + bl + xin @ Wr )
// Fused as [agg | xin] (M x 256) @ [Wl ; Wr] (256 x 128) via V_WMMA_F32_16X16X4_F32.
// Block = 8 waves; each wave computes one 16-row M-tile x all 128 cols.
__global__ __launch_bounds__(256) void k_sage_gemm(
    const float* __restrict__ summed, const float* __restrict__ invc,
    const float* __restrict__ xin,
    const float* __restrict__ Wl, const float* __restrict__ bl,
    const float* __restrict__ Wr,
    float* __restrict__ out, int relu)
{
  __shared__ __align__(16) float Wt[FDIM * WT_PITCH]; // Wt[n][k] = Wcat[k][n]

  const int tid = threadIdx.x;

  // Stage fused weight transposed into LDS. Consecutive threads read
  // consecutive n (coalesced global); LDS write banks (4n+k)%64.
  for (int i = tid; i < FDIM * KTOT; i += 256) {
    int n = i & (FDIM - 1);
    int k = i >> 7;
    float w = (k < FDIM) ? Wl[k * FDIM + n] : Wr[(k - FDIM) * FDIM + n];
    Wt[n * WT_PITCH + k] = w;
  }
  __syncthreads();

  int mtile = blockIdx.x * 8 + (tid >> 5);
  if (mtile >= MTILES) return;                 // uniform per wave: EXEC stays full

  const int lane = tid & 31;
  const int idx  = lane & 15;                  // A: row M ; B/D: col N (within tile)
  const int half = lane >> 4;                  // selects K pair / M offset

  const int arow = mtile * 16 + idx;
  const float rinv = invc[arow];
  const float* aggp = summed + (size_t)arow * FDIM + 2 * half;
  const float* xp   = xin    + (size_t)arow * FDIM + 2 * half;

  v8f acc[8];
#pragma unroll
  for (int n = 0; n < 8; ++n) acc[n] = (v8f){0.f,0.f,0.f,0.f,0.f,0.f,0.f,0.f};

  // --- K half 1: mean-aggregated neighbors (agg = summed * invc) @ Wl ---
#pragma unroll 2
  for (int kt = 0; kt < FDIM / 4; ++kt) {
    const int kb = kt * 4;
    v2f a = *(const v2f*)(aggp + kb);
    a = a * rinv;                              // fold scatter-mean normalization
#pragma unroll
    for (int n = 0; n < 8; ++n) {
      v2f b = *(const v2f*)(&Wt[(n * 16 + idx) * WT_PITCH + kb + 2 * half]);
      acc[n] = __builtin_amdgcn_wmma_f32_16x16x4_f32(
          false, a, false, b, (short)0, acc[n], false, false);
    }
  }

  // --- K half 2: root features xin @ Wr (Wt columns 128..255) ---
#pragma unroll 2
  for (int kt = 0; kt < FDIM / 4; ++kt) {
    const int kb = kt * 4;
    v2f a = *(const v2f*)(xp + kb);
#pragma unroll
    for (int n = 0; n < 8; ++n) {
      v2f b = *(const v2f*)(&Wt[(n * 16 + idx) * WT_PITCH + FDIM + kb + 2 * half]);
      acc[n] = __builtin_amdgcn_wmma_f32_16x16x4_f32(
          false, a, false, b, (short)0, acc[n], false, false);
    }
  }

  // Epilogue: D element (M = r + 8*half, N = idx) per VGPR r; add bias, ReLU.
  const int orow = mtile * 16 + half * 8;
#pragma unroll
  for (int n = 0; n < 8; ++n) {
    const float bb = bl[n * 16 + idx];
#pragma unroll
    for (int r = 0; r < 8; ++r) {
      float v = acc[n][r] + bb;
      if (relu) v = fmaxf(v, 0.f);
      out[(size_t)(orow + r) * FDIM + n * 16 + idx] = v;
    }
  }
}

extern "C" void kernel_launch(void* const* d_in, const int* in_sizes, int n_in,
                              void* d_out, int out_size, void* d_ws, size_t ws_size,
                              hipStream_t stream) {
  const float* x  = (const float*)d_in[0];
  const int*   ei = (const int*)d_in[1];
  const int* src = ei;
  const int* dst = ei + NE;
  const float* Wl[3] = {(const float*)d_in[2], (const float*)d_in[5], (const float*)d_in[8]};
  const float* bl[3] = {(const float*)d_in[3], (const float*)d_in[6], (const float*)d_in[9]};
  const float* Wr[3] = {(const float*)d_in[4], (const float*)d_in[7], (const float*)d_in[10]};

  float* ws     = (float*)d_ws;
  float* summed = ws;                               // NN*FDIM
  float* h      = ws + (size_t)NN * FDIM;           // NN*FDIM (in-place safe: each
                                                    // wave reads only its own rows
                                                    // before its final stores)
  float* cnt    = ws + (size_t)2 * NN * FDIM;       // NN
  float* invc   = cnt + NN;                         // NN
  // total ws: (2*6,400,000 + 100,000) * 4 B ≈ 51.6 MB

  // Degree counts once (edge list is identical for all layers)
  k_zero4<<<(NN / 4 + 255) / 256, 256, 0, stream>>>((float4*)cnt, NN / 4);
  k_count<<<(NE + 255) / 256, 256, 0, stream>>>(dst, cnt);
  k_invc<<<(NN + 255) / 256, 256, 0, stream>>>(cnt, invc);

  const float* xin = x;
  for (int l = 0; l < 3; ++l) {
    float* xout = (l == 2) ? (float*)d_out : h;
    k_zero4<<<((NN * FDIM / 4) + 255) / 256, 256, 0, stream>>>((float4*)summed, NN * FDIM / 4);
    k_scatter<<<NE / 8, 256, 0, stream>>>(src, dst, xin, summed);
    k_sage_gemm<<<(MTILES + 7) / 8, 256, 0, stream>>>(summed, invc, xin,
                                                      Wl[l], bl[l], Wr[l],
                                                      xout, (l < 2) ? 1 : 0);
    xin = xout;
  }
}